// RenderNet_72962904425042
// MI455X (gfx1250) — compile-verified
//
#include <hip/hip_runtime.h>
#include <hip/hip_bf16.h>
#include <cstdint>
#include <math.h>

// Problem constants (fixed by the reference setup).
#define NPIX   65536   // Y*X rays
#define C_CH   32
#define ZS     64
#define TILE_PX 256    // pixels per block
#define HALF_Z 32      // z-rows per TDM tile (two tiles per channel)

typedef __attribute__((ext_vector_type(4))) unsigned int tdm_v4u;
typedef __attribute__((ext_vector_type(8))) int          tdm_v8i;
typedef __attribute__((ext_vector_type(4))) int          tdm_v4i;

// Issue one TDM DMA: feat tile t (t = c*2 + h) -> LDS at lds_byte_off.
// Tile = HALF_Z rows x TILE_PX elements, row stride NPIX elements, FP32.
__device__ __forceinline__ void tdm_issue_tile(const float* feat,
                                               unsigned lds_byte_off,
                                               int t, int pbase) {
  unsigned long long ga = (unsigned long long)(uintptr_t)feat
                        + (unsigned long long)t * (unsigned long long)(HALF_Z * NPIX) * 4ull
                        + (unsigned long long)pbase * 4ull;
  // D# group 0: count=1 | lds_addr | global_addr[56:0] | type=2
  tdm_v4u g0 = { 1u,
                 lds_byte_off,
                 (unsigned)(ga & 0xFFFFFFFFull),
                 (unsigned)((ga >> 32) & 0x01FFFFFFull) | (2u << 30) };
  const unsigned td0 = NPIX;        // tensor_dim0 (elements) -- large enough: no OOB
  const unsigned td1 = C_CH * ZS;   // tensor_dim1
  // D# group 1: data_size=4B(code 2); tile 256 x 32; dim0_stride = 65536 elements
  tdm_v8i g1 = {
      (int)(2u << 16),                                            // data_size=2 (4B)
      (int)((td0 & 0xFFFFu) << 16),                               // abar=0 | td0.lo16
      (int)(((td0 >> 16) & 0xFFFFu) | ((td1 & 0xFFFFu) << 16)),   // td0.hi16 | td1.lo16
      (int)(((td1 >> 16) & 0xFFFFu) | ((unsigned)TILE_PX << 16)), // td1.hi16 | tile_dim0
      (int)HALF_Z,                                                // tile_dim1 | tile_dim2=0
      (int)NPIX,                                                  // dim0_stride lo32
      0,                                                          // stride hi16 | dim1_stride lo16
      0 };
  tdm_v4i gz4 = { 0, 0, 0, 0 };                    // groups 2/3: unused (2D tensor)
  tdm_v8i gz8 = { 0, 0, 0, 0, 0, 0, 0, 0 };        // extra group (clang-23 6-arg form)
  __builtin_amdgcn_tensor_load_to_lds(g0, g1, gz4, gz4, gz8, 0);
}

__global__ __launch_bounds__(TILE_PX)
void rendernet_fused(const float* __restrict__ feat,
                     const float* __restrict__ occ,
                     const float* __restrict__ zd,
                     const float* __restrict__ noise,
                     float* __restrict__ out) {
  extern __shared__ float sfeat[];              // 2 * HALF_Z * TILE_PX floats = 64 KB
  const int tid   = threadIdx.x;
  const int pbase = blockIdx.x * TILE_PX;
  const int n     = pbase + tid;

  const unsigned lds_base = (unsigned)(uintptr_t)(void*)&sfeat[0];

  // Kick off DMA of tile 0 before the (long) weight computation -> overlap.
  if (tid < 32) {
    tdm_issue_tile(feat, lds_base, 0, pbase);
  }
  if (n == 0) out[0] = 0.0f;                    // total_loss == 0

  // ---------- per-ray weights: alpha / cumprod transmittance (FP32) ----------
  float w[ZS];
  const float4* nz4 = reinterpret_cast<const float4*>(noise + (size_t)n * ZS);
  float T = 1.0f;
  float wsum = 0.0f;
#pragma unroll
  for (int q = 0; q < ZS / 4; ++q) {
    float4 nv = nz4[q];
    float nn[4] = { nv.x, nv.y, nv.z, nv.w };
#pragma unroll
    for (int j = 0; j < 4; ++j) {
      const int z   = q * 4 + j;
      const float o = occ[(size_t)z * NPIX + n];           // coalesced per z-plane
      const float d = (z < ZS - 1) ? (zd[z + 1] - zd[z]) : 1e10f;
      float raw = o + nn[j] * 0.1f;
      raw = raw > 0.0f ? raw : 0.0f;                       // relu
      const float alpha = 1.0f - expf(-raw * d);
      const float wz = alpha * T;
      w[z]  = wz;
      wsum += wz;
      T *= (1.0f - alpha + 1e-10f);                        // cumprod step
    }
  }
  float accv = wsum < 0.0f ? 0.0f : (wsum > 1.0f ? 1.0f : wsum);
  out[1 + 2 * (C_CH * NPIX) + n] = accv;                   // acc_map

  // ---------- channel composite from TDM-staged, double-buffered LDS ----------
  for (int c = 0; c < C_CH; ++c) {
    float facc = 0.0f;
#pragma unroll
    for (int h = 0; h < 2; ++h) {
      const int t = c * 2 + h;
      if (tid < 32) {                                      // wave 0 drives the DMA
        if (t + 1 < 2 * C_CH) {
          tdm_issue_tile(feat,
                         lds_base + (unsigned)((t + 1) & 1) * (HALF_Z * TILE_PX * 4u),
                         t + 1, pbase);
          __builtin_amdgcn_s_wait_tensorcnt(1);            // in-order: tile t done
        } else {
          __builtin_amdgcn_s_wait_tensorcnt(0);
        }
      }
      __syncthreads();                                     // publish tile t to all waves
      const float* fb = sfeat + (t & 1) * (HALF_Z * TILE_PX);
#pragma unroll
      for (int zz = 0; zz < HALF_Z; ++zz) {
        facc = fmaf(w[h * HALF_Z + zz], fb[zz * TILE_PX + tid], facc);
      }
      __syncthreads();                                     // buffer free for prefetch t+2
    }
    const float sg = 1.0f / (1.0f + expf(-facc)) - 0.5f;   // sigmoid - 0.5
    out[1 + c * NPIX + n]                = sg;             // rgb_e
    out[1 + C_CH * NPIX + c * NPIX + n]  = facc;           // feat_e
  }
}

extern "C" void kernel_launch(void* const* d_in, const int* in_sizes, int n_in,
                              void* d_out, int out_size, void* d_ws, size_t ws_size,
                              hipStream_t stream) {
  (void)in_sizes; (void)n_in; (void)out_size; (void)d_ws; (void)ws_size;
  const float* feat  = (const float*)d_in[0];   // [1,32,64,256,256]
  const float* occ   = (const float*)d_in[1];   // [1,1,64,256,256]
  const float* zdist = (const float*)d_in[2];   // [64]
  const float* noise = (const float*)d_in[3];   // [65536,64]
  float* out = (float*)d_out;                   // [1 | 2097152 | 2097152 | 65536]

  const size_t shmem = (size_t)2 * HALF_Z * TILE_PX * sizeof(float);  // 64 KB
  hipLaunchKernelGGL(rendernet_fused,
                     dim3(NPIX / TILE_PX), dim3(TILE_PX), shmem, stream,
                     feat, occ, zdist, noise, out);
}